// CatGCNEncoder_27367531610202
// MI455X (gfx1250) — compile-verified
//
#include <hip/hip_runtime.h>

typedef __attribute__((ext_vector_type(2))) float v2f;
typedef __attribute__((ext_vector_type(8))) float v8f;

#define NU   100000
#define NI   50000
#define DD   512
#define OO   500
#define SS   5
#define EE   200000
#define SUB  100

// ---------------------------------------------------------------------------
// Dense GEMM via V_WMMA_F32_16X16X4_F32.
// Block: 256 threads = 8 waves (4 row-groups x 2 col-groups) -> 128x128 tile.
// Wave: 32 rows x 64 cols = 2 A-frags x 4 B-frags = 8 WMMAs per K-step.
// B fragments are loaded into 4 independent values per K-step so the backend
// can keep multiple DS loads in flight (partial s_wait_dscnt) instead of
// serializing every WMMA pair on one register pair.
// Output written directly into [S, nrows, SUB] layout (col j -> s=j/100, k=j%100).
// ---------------------------------------------------------------------------
__global__ __launch_bounds__(256)
void gemm_wmma(const float* __restrict__ A, const float* __restrict__ W,
               float* __restrict__ tmp, int nrows)
{
    __shared__ float ldsA[128 * 36];         // 18.4 KB: 128 rows x 32 k, pad 36
    __shared__ float ldsWc[8 * 32 * 16];     // 16.4 KB: 8 col-tiles x 32 k x 16 j

    const int tid  = threadIdx.x;
    const int wave = tid >> 5;
    const int lane = tid & 31;
    const int half = lane >> 4;              // 0: lanes 0-15, 1: lanes 16-31
    const int l16  = lane & 15;

    const int wr = wave & 3;                 // row-group 0..3  (32 rows each)
    const int wc = wave >> 2;                // col-group 0..1  (64 cols each)

    const int rowBase = blockIdx.x * 128;
    const int colBase = blockIdx.y * 128;

    v8f acc[2][4] = {};

    for (int kc = 0; kc < DD; kc += 32) {
        __syncthreads();

        // Stage A chunk [128 rows x 32 k] as float4 (coalesced; rows clamped).
        for (int idx = tid; idx < 128 * 8; idx += 256) {
            const int r  = idx >> 3;
            const int kq = idx & 7;
            int gr = rowBase + r;
            if (gr >= nrows) gr = nrows - 1;
            const float4 v = *(const float4*)(A + (size_t)gr * DD + kc + kq * 4);
            *(float4*)(&ldsA[r * 36 + kq * 4]) = v;
        }

        // Stage W chunk [32 k x 128 cols] tile-major (zero-fill past O).
        for (int idx = tid; idx < 32 * 128; idx += 256) {
            const int c  = idx & 127;
            const int k  = idx >> 7;
            const int gj = colBase + c;
            const float w = (gj < OO) ? W[(size_t)(kc + k) * OO + gj] : 0.0f;
            ldsWc[(c >> 4) * 512 + k * 16 + (c & 15)] = w;
        }
        __syncthreads();

        #pragma unroll
        for (int kk = 0; kk < 32; kk += 4) {
            // A fragments: lane holds A[M=l16][K = kk+2*half, +1] for both row-groups
            v2f a0 = *(const v2f*)(&ldsA[(wr * 32 +  0 + l16) * 36 + kk + 2 * half]);
            v2f a1 = *(const v2f*)(&ldsA[(wr * 32 + 16 + l16) * 36 + kk + 2 * half]);

            // 4 independent B fragments: lane holds W[K][N=l16], K = kk+2*half, +1
            v2f b[4];
            #pragma unroll
            for (int t = 0; t < 4; ++t) {
                const float* bp =
                    &ldsWc[(wc * 4 + t) * 512 + (kk + 2 * half) * 16 + l16];
                b[t].x = bp[0];
                b[t].y = bp[16];   // next k row
            }

            #pragma unroll
            for (int t = 0; t < 4; ++t) {
                acc[0][t] = __builtin_amdgcn_wmma_f32_16x16x4_f32(
                    false, a0, false, b[t], (short)0, acc[0][t], false, false);
                acc[1][t] = __builtin_amdgcn_wmma_f32_16x16x4_f32(
                    false, a1, false, b[t], (short)0, acc[1][t], false, false);
            }
        }
    }

    // Epilogue: C VGPR r -> M = r + 8*half, N = l16.
    #pragma unroll
    for (int t = 0; t < 4; ++t) {
        const int j = colBase + wc * 64 + t * 16 + l16;
        if (j < OO) {
            const int s  = j / SUB;
            const int kk = j - s * SUB;
            #pragma unroll
            for (int rg = 0; rg < 2; ++rg) {
                #pragma unroll
                for (int r = 0; r < 8; ++r) {
                    const int row = rowBase + wr * 32 + rg * 16 + r + 8 * half;
                    if (row < nrows) {
                        tmp[(size_t)s * nrows * SUB + (size_t)row * SUB + kk] =
                            acc[rg][t][r];
                    }
                }
            }
        }
    }
}

// ---------------------------------------------------------------------------
// COO SpMM scatter: one wave per edge. 25 lanes gather one float4 each of the
// 400 B source row (L2-resident) and atomically add into the dest row segment.
// ---------------------------------------------------------------------------
__global__ __launch_bounds__(256)
void spmm_scatter(const int* __restrict__ rows, const int* __restrict__ cols,
                  const float* __restrict__ vals, const float* __restrict__ tmp,
                  float* __restrict__ out, int nsrc)
{
    const int wid  = (blockIdx.x * blockDim.x + threadIdx.x) >> 5;
    const int lane = threadIdx.x & 31;
    if (wid >= SS * EE) return;

    const int   s = wid / EE;
    const int   r = rows[wid];
    const int   c = cols[wid];
    const float v = vals[wid];

    const float4* src = (const float4*)(tmp + ((size_t)s * nsrc + c) * SUB);
    float*        dst = out + (size_t)r * OO + s * SUB;

    if (lane < SUB / 4) {
        const float4 t = src[lane];
        atomicAdd(dst + lane * 4 + 0, v * t.x);
        atomicAdd(dst + lane * 4 + 1, v * t.y);
        atomicAdd(dst + lane * 4 + 2, v * t.z);
        atomicAdd(dst + lane * 4 + 3, v * t.w);
    }
}

// ---------------------------------------------------------------------------
// Vectorized ReLU over the whole output (75M floats, divisible by 4).
// ---------------------------------------------------------------------------
__global__ __launch_bounds__(256)
void relu_kernel(float4* __restrict__ out, int n4)
{
    const int i = blockIdx.x * blockDim.x + threadIdx.x;
    if (i < n4) {
        float4 v = out[i];
        v.x = fmaxf(v.x, 0.0f);
        v.y = fmaxf(v.y, 0.0f);
        v.z = fmaxf(v.z, 0.0f);
        v.w = fmaxf(v.w, 0.0f);
        out[i] = v;
    }
}

extern "C" void kernel_launch(void* const* d_in, const int* in_sizes, int n_in,
                              void* d_out, int out_size, void* d_ws, size_t ws_size,
                              hipStream_t stream)
{
    const float* user_in = (const float*)d_in[0];
    const float* item_in = (const float*)d_in[1];
    const float* weight  = (const float*)d_in[2];
    const int*   u_rows  = (const int*)d_in[3];
    const int*   u_cols  = (const int*)d_in[4];
    const float* u_vals  = (const float*)d_in[5];
    const int*   i_rows  = (const int*)d_in[6];
    const int*   i_cols  = (const int*)d_in[7];
    const float* i_vals  = (const float*)d_in[8];
    float*       out     = (float*)d_out;

    // Workspace: tmp_u [S,NU,SUB] (200 MB) then tmp_v [S,NI,SUB] (100 MB).
    float* tmp_u = (float*)d_ws;
    float* tmp_v = tmp_u + (size_t)SS * NU * SUB;

    hipMemsetAsync(d_out, 0, (size_t)out_size * sizeof(float), stream);

    const dim3 blk(256);
    const dim3 gU((NU + 127) / 128, (OO + 127) / 128);
    gemm_wmma<<<gU, blk, 0, stream>>>(user_in, weight, tmp_u, NU);
    const dim3 gI((NI + 127) / 128, (OO + 127) / 128);
    gemm_wmma<<<gI, blk, 0, stream>>>(item_in, weight, tmp_v, NI);

    const int nWaves  = SS * EE;                       // 1M edges per side
    const int nBlocks = (nWaves * 32 + 255) / 256;     // 8 waves / block
    // user_h: aggregate item-side features (tmp_v) into user rows
    spmm_scatter<<<nBlocks, blk, 0, stream>>>(u_rows, u_cols, u_vals, tmp_v, out, NI);
    // item_h: aggregate user-side features (tmp_u) into item rows
    spmm_scatter<<<nBlocks, blk, 0, stream>>>(i_rows, i_cols, i_vals, tmp_u,
                                              out + (size_t)NU * OO, NU);

    const int n4 = out_size / 4;
    relu_kernel<<<(n4 + 255) / 256, blk, 0, stream>>>((float4*)d_out, n4);
}